// Encoder_82205674045926
// MI455X (gfx1250) — compile-verified
//
#include <hip/hip_runtime.h>
#include <stdint.h>

typedef __attribute__((ext_vector_type(2))) float v2f;
typedef __attribute__((ext_vector_type(8))) float v8f;

#define D 128  // feature dim for all layers

// ---------------------------------------------------------------------------
// Utility kernels
// ---------------------------------------------------------------------------
__global__ __launch_bounds__(256) void zero_f32(float* __restrict__ p, int n) {
    int i = blockIdx.x * blockDim.x + threadIdx.x;
    if (i < n) p[i] = 0.0f;
}

__global__ __launch_bounds__(256) void copy_f32x4(const float4* __restrict__ in,
                                                  float4* __restrict__ out, int n4) {
    int i = blockIdx.x * blockDim.x + threadIdx.x;
    if (i < n4) out[i] = in[i];
}

// ---------------------------------------------------------------------------
// Edge scatter-add: agg[dst] += h[src]  (+ deg[dst] += 1 on first pass)
// One wave (32 lanes) per edge; each lane moves one float4 (128/32 floats).
// Roofline: 102.4M global_atomic_add_f32 per layer into a 25.6MB table,
// fully L2-resident (192MB L2) -> bound by L2 atomic throughput, not HBM.
// ---------------------------------------------------------------------------
__global__ __launch_bounds__(256) void sage_scatter(
    const int* __restrict__ src, const int* __restrict__ dst,
    const float* __restrict__ hin, float* __restrict__ agg,
    float* __restrict__ deg, int n_edges, int add_deg)
{
    long long gid = (long long)blockIdx.x * blockDim.x + threadIdx.x;
    int lane      = (int)(gid & 31);
    long long e   = gid >> 5;
    if (e >= n_edges) return;

    int s = src[e];
    int d = dst[e];

    const float4 v = ((const float4*)(hin + (long long)s * D))[lane];
    float* ap = agg + (long long)d * D + lane * 4;
    atomicAdd(ap + 0, v.x);
    atomicAdd(ap + 1, v.y);
    atomicAdd(ap + 2, v.z);
    atomicAdd(ap + 3, v.w);

    if (add_deg && lane == 0) atomicAdd(deg + d, 1.0f);
}

// ---------------------------------------------------------------------------
// Fused GEMM + per-row scale + bias (+ReLU):
//   out[n, :] = relu?( (agg[n,:] @ W^T) / (deg[n]+1) + b )
// (normalization commutes with the row-wise GEMM, so it moves to the epilogue)
//
// Block = 256 threads = 8 waves; block covers 16 nodes x 128 cols, wave w owns
// the 16x16 tile for cols [16w,16w+16). K-loop: 32x V_WMMA_F32_16X16X4_F32.
// A tile (16x128 f32, row-padded) staged global->LDS with
// GLOBAL_LOAD_ASYNC_TO_LDS_B128 (ASYNCcnt), then ds_load'ed per WMMA layout.
//
// VGPR layouts (CDNA5 ISA 7.12.2):
//   A 16x4 f32: M = lane&15; v0 = K(2*half), v1 = K(2*half+1)
//   B 4x16 f32 (mirrors A/CD striping): N = lane&15; v0/v1 same K pairing
//   C/D 16x16 f32: VGPR r -> M=r (lanes 0-15) / M=r+8 (lanes 16-31), N=lane&15
// ---------------------------------------------------------------------------
template <int RELU>
__global__ __launch_bounds__(256) void sage_gemm(
    const float* __restrict__ agg, const float* __restrict__ deg,
    const float* __restrict__ W, const float* __restrict__ bias,
    float* __restrict__ out, int n_nodes)
{
    __shared__ float sh[16][D + 4];  // +4 float pad: lanes hit distinct banks
    __shared__ float s_inv[16];      // 1/(deg+1) per tile row

    const int tid  = threadIdx.x;
    const int row0 = blockIdx.x * 16;
    const bool full_tile = (row0 + 16 <= n_nodes);

    // Per-row reciprocal of (deg+1)
    if (tid < 16) {
        int node = row0 + tid;
        float dv = (node < n_nodes) ? deg[node] : 0.0f;
        s_inv[tid] = 1.0f / (dv + 1.0f);
    }

    // Stage raw agg tile (16 rows x 128 f32) into padded LDS rows using the
    // CDNA5 async global->LDS DMA path. 512 chunks of 16B; 2 per thread.
    for (int i = tid; i < 16 * (D / 4); i += 256) {
        int r  = i >> 5;        // row 0..15
        int c4 = i & 31;        // 16B chunk within row
        int node = row0 + r;
        if (node < n_nodes) {
            unsigned lds_byte = (unsigned)(uintptr_t)(&sh[r][c4 * 4]);
            const float* gp = agg + (size_t)node * D + c4 * 4;
            asm volatile("global_load_async_to_lds_b128 %0, %1, off"
                         :: "v"(lds_byte), "v"(gp)
                         : "memory");
        } else {
            float4 z = {0.0f, 0.0f, 0.0f, 0.0f};
            *(float4*)&sh[r][c4 * 4] = z;
        }
    }
    asm volatile("s_wait_asynccnt 0" ::: "memory");
    __syncthreads();

    const int wave = tid >> 5;
    const int lane = tid & 31;
    const int half = lane >> 4;   // 0: lanes 0-15, 1: lanes 16-31
    const int l16  = lane & 15;
    const int oc   = wave * 16 + l16;   // this lane's output column for B/D

    __builtin_prefetch(W + oc * D, 0, 0);   // pull this lane's W row toward L0

    v8f acc = {};
#pragma unroll 4
    for (int kk = 0; kk < D; kk += 4) {
        const int k0 = kk + 2 * half;
        v2f a, b;
        a.x = sh[l16][k0];
        a.y = sh[l16][k0 + 1];
        // B = W^T : B[k][oc] = W[oc][k]; W is 64KB -> L2/L0 resident
        b.x = W[oc * D + k0];
        b.y = W[oc * D + k0 + 1];
        acc = __builtin_amdgcn_wmma_f32_16x16x4_f32(
            /*neg_a=*/false, a, /*neg_b=*/false, b,
            /*c_mod=*/(short)0, acc, /*reuse_a=*/false, /*reuse_b=*/false);
    }

    const float bv = bias[oc];
    if (full_tile) {
        // Uniform fast path: no per-row exec masking.
#pragma unroll
        for (int r = 0; r < 8; ++r) {
            int node = row0 + r + half * 8;       // C/D: M = r + 8*half
            float v  = acc[r] * s_inv[r + half * 8] + bv;
            if (RELU) v = fmaxf(v, 0.0f);
            out[(size_t)node * D + oc] = v;
        }
    } else {
#pragma unroll
        for (int r = 0; r < 8; ++r) {
            int node = row0 + r + half * 8;
            float v  = acc[r] * s_inv[r + half * 8] + bv;
            if (RELU) v = fmaxf(v, 0.0f);
            if (node < n_nodes) out[(size_t)node * D + oc] = v;
        }
    }
}

// ---------------------------------------------------------------------------
// Launch: 2-layer GraphSAGE (gcn aggregator)
// ---------------------------------------------------------------------------
extern "C" void kernel_launch(void* const* d_in, const int* in_sizes, int n_in,
                              void* d_out, int out_size, void* d_ws, size_t ws_size,
                              hipStream_t stream)
{
    const float* h  = (const float*)d_in[0];
    const int*   ei = (const int*)d_in[1];
    const float* W1 = (const float*)d_in[2];
    const float* b1 = (const float*)d_in[3];
    const float* W2 = (const float*)d_in[4];
    const float* b2 = (const float*)d_in[5];
    float* out = (float*)d_out;

    const int n_nodes = in_sizes[0] / D;   // 50000
    const int n_edges = in_sizes[1] / 2;   // 800000
    const int* src = ei;
    const int* dst = ei + n_edges;

    // Workspace layout (floats): deg | agg | h1   (~51.5 MB total)
    float* deg = (float*)d_ws;
    float* agg = deg + (((size_t)n_nodes + 63) & ~(size_t)63);
    float* h1  = agg + (size_t)n_nodes * D;

    const int feat4 = n_nodes * D / 4;                    // float4 count
    dim3 blk(256);
    dim3 g_zero((n_nodes + 255) / 256);
    dim3 g_copy((feat4 + 255) / 256);
    dim3 g_scat((int)(((long long)n_edges * 32 + 255) / 256));
    dim3 g_gemm((n_nodes + 15) / 16);

    // ---- Layer 1 ----
    zero_f32<<<g_zero, blk, 0, stream>>>(deg, n_nodes);
    copy_f32x4<<<g_copy, blk, 0, stream>>>((const float4*)h, (float4*)agg, feat4);
    sage_scatter<<<g_scat, blk, 0, stream>>>(src, dst, h, agg, deg, n_edges, /*add_deg=*/1);
    sage_gemm<1><<<g_gemm, blk, 0, stream>>>(agg, deg, W1, b1, h1, n_nodes);

    // ---- Layer 2 ----
    copy_f32x4<<<g_copy, blk, 0, stream>>>((const float4*)h1, (float4*)agg, feat4);
    sage_scatter<<<g_scat, blk, 0, stream>>>(src, dst, h1, agg, deg, n_edges, /*add_deg=*/0);
    sage_gemm<0><<<g_gemm, blk, 0, stream>>>(agg, deg, W2, b2, out, n_nodes);
}